// VMEncoder_28544352649753
// MI455X (gfx1250) — compile-verified
//
#include <hip/hip_runtime.h>
#include <stdint.h>

// ---------------------------------------------------------------------------
// VM (TensoRF) encoder for MI455X / gfx1250.
//   x      : (N, 3) fp32 in [-1,1]
//   C_mat  : (3, 32, 512, 512) fp32 planes
//   C_vec  : (3, 32, 512, 1)   fp32 lines
//   out    : (N, 96) fp32, out[p, i*32+c] = bilerp(plane_i)[c] * lerp(line_i)[c]
// Strategy:
//   pass 1: transpose planes to channel-last (3,512,512,32) in workspace
//           (one-time 192 MB HBM traffic; table then lives in the 192 MB L2)
//   pass 2: transpose lines to (3,512,32) in workspace (192 KB)
//   pass 3: main kernel, 1024-thr blocks (32 waves = full WGP at 192 KB LDS).
//           Wave 0 TDM-loads the whole transposed line table into LDS with one
//           tensor_load_to_lds, s_wait_tensorcnt, barrier. Grid-stride loop:
//           each point does 4 corners x 8 b128 gathers per plane from L2,
//           2 taps x 8 ds_load_b128 from LDS, non-temporal b128 stores.
// ---------------------------------------------------------------------------

typedef float   f4  __attribute__((ext_vector_type(4)));
typedef unsigned int v4u __attribute__((ext_vector_type(4)));
typedef int     v8i __attribute__((ext_vector_type(8)));
typedef int     v4i __attribute__((ext_vector_type(4)));

#define RES   512
#define COUT  32
#define VEC_ELEMS (3 * RES * COUT)            // 49152 floats = 192 KB
#define MATT_ELEMS ((size_t)3 * RES * RES * COUT)
#define MAIN_BLOCKS 512                       // grid-stride: halves TDM restaging

__device__ __forceinline__ int iclamp(int v, int lo, int hi) {
    return v < lo ? lo : (v > hi ? hi : v);
}

// ---- pass 1: (3,32,512,512) -> (3,512,512,32), LDS 64x32 tile transpose ----
__global__ __launch_bounds__(256) void k_transpose_mat(const float* __restrict__ in,
                                                       float* __restrict__ outp) {
    __shared__ float tile[64][33];            // +1 pad: conflict-free both phases
    int t  = blockIdx.x;                      // 3*512*8 blocks
    int xt = t & 7;                           // x-tile (64 wide)
    int iy = t >> 3;                          // i*512 + y
    int y  = iy & (RES - 1);
    int i  = iy >> 9;
    int x0 = xt * 64;
    int tid = threadIdx.x;

    int xl = tid & 63;                        // coalesced along x on loads
    int cb = tid >> 6;                        // 0..3
    const float* base = in + ((size_t)(i * COUT) * RES + y) * RES;
#pragma unroll
    for (int k = 0; k < 8; ++k) {
        int c = cb * 8 + k;
        tile[xl][c] = base[(size_t)c * RES * RES + x0 + xl];
    }
    __syncthreads();

    int cw = tid & 31;                        // coalesced along c on stores
    int xg = tid >> 5;                        // 0..7
    float* ob = outp + (((size_t)i * RES + y) * RES + x0) * COUT;
#pragma unroll
    for (int k = 0; k < 8; ++k) {
        int xv = xg * 8 + k;
        ob[(size_t)xv * COUT + cw] = tile[xv][cw];
    }
}

// ---- pass 2: (3,32,512) -> (3,512,32) ----
__global__ __launch_bounds__(256) void k_transpose_vec(const float* __restrict__ in,
                                                       float* __restrict__ outp) {
    int j = blockIdx.x * 256 + threadIdx.x;
    if (j >= VEC_ELEMS) return;
    int i = j / (RES * COUT);
    int r = j % (RES * COUT);
    int y = r / COUT;
    int c = r % COUT;
    outp[j] = in[((size_t)(i * COUT + c)) * RES + y];
}

// ---- pass 3: main gather kernel (channel-last table + TDM-staged LDS vec) ----
__global__ __launch_bounds__(1024, 1) void k_vm_main(const float* __restrict__ xin,
                                                     const float* __restrict__ matT,
                                                     const float* __restrict__ vecT,
                                                     float* __restrict__ outp, int N) {
    extern __shared__ float svec[];           // 3*512*32 floats, layout [i][y][c]

    // Wave 0 issues one TDM load of the whole transposed line table.
    if (threadIdx.x < 32) {
        uint32_t lds = (uint32_t)(uintptr_t)(void*)svec;   // low 32 bits = LDS byte addr
        uint64_t ga  = (uint64_t)(uintptr_t)vecT;
        v4u g0;
        g0.x = 1u;                                          // count=1, user descriptor
        g0.y = lds;                                         // lds_addr [63:32]
        g0.z = (uint32_t)ga;                                // global_addr [95:64]
        g0.w = (uint32_t)((ga >> 32) & 0x1FFFFFFull)        // global_addr [120:96]
             | (2u << 30);                                  // type=2 ("image")
        v8i g1;
        g1[0] = (int)(2u << 16);        // workgroup_mask=0, data_size=2 (4 B)
        g1[1] = (int)(32u << 16);       // tensor_dim0 = 32 (contiguous channels)
        g1[2] = (int)(1536u << 16);     // tensor_dim0 hi=0 | tensor_dim1 = 1536 rows
        g1[3] = (int)(32u << 16);       // tensor_dim1 hi=0 | tile_dim0 = 32
        g1[4] = 1536;                   // tile_dim1 = 1536 (whole tensor)
        g1[5] = 32;                     // tensor_dim0_stride = 32
        g1[6] = 0;
        g1[7] = 0;
        v4i z4 = {0, 0, 0, 0};          // groups 2/3 unused (2-D tensor)
        v8i z8 = {0, 0, 0, 0, 0, 0, 0, 0};
        __builtin_amdgcn_tensor_load_to_lds(g0, g1, z4, z4, z8, 0);
        __builtin_amdgcn_s_wait_tensorcnt(0);
    }
    __syncthreads();

    for (int p = blockIdx.x * 1024 + threadIdx.x; p < N; p += gridDim.x * 1024) {
        // streamed once: non-temporal so coords don't evict the L2 plane table
        float c0 = __builtin_nontemporal_load(xin + 3 * p + 0);
        float c1 = __builtin_nontemporal_load(xin + 3 * p + 1);
        float c2 = __builtin_nontemporal_load(xin + 3 * p + 2);
        // MAT_IDS = {{0,1},{0,2},{1,2}} (x=width, y=height); VEC_IDS = {2,1,0}
        const float AX[3] = {c0, c0, c1};
        const float AY[3] = {c1, c2, c2};
        const float VY[3] = {c2, c1, c0};

        f4* op = (f4*)(outp + (size_t)p * 96);

#pragma unroll
        for (int i = 0; i < 3; ++i) {
            // align_corners=False: ix = ((x+1)*512 - 1)*0.5 = x*256 + 255.5
            float ix = fmaf(AX[i], 256.f, 255.5f);
            float iy = fmaf(AY[i], 256.f, 255.5f);
            float fx = floorf(ix), fy = floorf(iy);
            float wx = ix - fx,    wy = iy - fy;
            int x0 = (int)fx, y0 = (int)fy;
            int x1 = x0 + 1,  y1 = y0 + 1;
            float vx0 = ((unsigned)x0 < (unsigned)RES) ? 1.f : 0.f;
            float vx1 = ((unsigned)x1 < (unsigned)RES) ? 1.f : 0.f;
            float vy0 = ((unsigned)y0 < (unsigned)RES) ? 1.f : 0.f;
            float vy1 = ((unsigned)y1 < (unsigned)RES) ? 1.f : 0.f;
            float w00 = (1.f - wx) * (1.f - wy) * vx0 * vy0;
            float w01 = wx * (1.f - wy) * vx1 * vy0;
            float w10 = (1.f - wx) * wy * vx0 * vy1;
            float w11 = wx * wy * vx1 * vy1;
            int xc0 = iclamp(x0, 0, RES - 1), xc1 = iclamp(x1, 0, RES - 1);
            int yc0 = iclamp(y0, 0, RES - 1), yc1 = iclamp(y1, 0, RES - 1);

            const float* g00p = matT + ((((size_t)i * RES + yc0) * RES + xc0) * COUT);
            const float* g01p = matT + ((((size_t)i * RES + yc0) * RES + xc1) * COUT);
            const float* g10p = matT + ((((size_t)i * RES + yc1) * RES + xc0) * COUT);
            const float* g11p = matT + ((((size_t)i * RES + yc1) * RES + xc1) * COUT);

            // line: grid is (C,512,1); x-coord=0 -> ix=0, wx=0 (x=1 tap weight 0)
            float ivy = fmaf(VY[i], 256.f, 255.5f);
            float fvy = floorf(ivy);
            float wv  = ivy - fvy;
            int yv0 = (int)fvy, yv1 = yv0 + 1;
            float w0v = (1.f - wv) * (((unsigned)yv0 < (unsigned)RES) ? 1.f : 0.f);
            float w1v = wv * (((unsigned)yv1 < (unsigned)RES) ? 1.f : 0.f);
            int yv0c = iclamp(yv0, 0, RES - 1), yv1c = iclamp(yv1, 0, RES - 1);
            const float* v0p = svec + ((size_t)i * RES + yv0c) * COUT;
            const float* v1p = svec + ((size_t)i * RES + yv1c) * COUT;

#pragma unroll
            for (int c = 0; c < COUT; c += 4) {
                f4 g00 = *(const f4*)(g00p + c);
                f4 g01 = *(const f4*)(g01p + c);
                f4 g10 = *(const f4*)(g10p + c);
                f4 g11 = *(const f4*)(g11p + c);
                f4 va  = *(const f4*)(v0p + c);     // ds_load_b128
                f4 vb  = *(const f4*)(v1p + c);
                f4 m, v, r;
                m.x = fmaf(w11, g11.x, fmaf(w10, g10.x, fmaf(w01, g01.x, w00 * g00.x)));
                m.y = fmaf(w11, g11.y, fmaf(w10, g10.y, fmaf(w01, g01.y, w00 * g00.y)));
                m.z = fmaf(w11, g11.z, fmaf(w10, g10.z, fmaf(w01, g01.z, w00 * g00.z)));
                m.w = fmaf(w11, g11.w, fmaf(w10, g10.w, fmaf(w01, g01.w, w00 * g00.w)));
                v.x = fmaf(w1v, vb.x, w0v * va.x);
                v.y = fmaf(w1v, vb.y, w0v * va.y);
                v.z = fmaf(w1v, vb.z, w0v * va.z);
                v.w = fmaf(w1v, vb.w, w0v * va.w);
                r.x = m.x * v.x; r.y = m.y * v.y; r.z = m.z * v.z; r.w = m.w * v.w;
                __builtin_nontemporal_store(r, op + (i * 8 + (c >> 2)));
            }
        }
    }
}

// ---- fallback: direct gathers from original layouts (if workspace too small) ----
__global__ __launch_bounds__(256) void k_vm_direct(const float* __restrict__ xin,
                                                   const float* __restrict__ Cm,
                                                   const float* __restrict__ Cv,
                                                   float* __restrict__ outp, int N) {
    int p = blockIdx.x * 256 + threadIdx.x;
    if (p >= N) return;
    float c0 = xin[3 * p + 0], c1 = xin[3 * p + 1], c2 = xin[3 * p + 2];
    const float AX[3] = {c0, c0, c1};
    const float AY[3] = {c1, c2, c2};
    const float VY[3] = {c2, c1, c0};
#pragma unroll
    for (int i = 0; i < 3; ++i) {
        float ix = fmaf(AX[i], 256.f, 255.5f);
        float iy = fmaf(AY[i], 256.f, 255.5f);
        float fx = floorf(ix), fy = floorf(iy);
        float wx = ix - fx, wy = iy - fy;
        int x0 = (int)fx, y0 = (int)fy, x1 = x0 + 1, y1 = y0 + 1;
        float vx0 = ((unsigned)x0 < (unsigned)RES) ? 1.f : 0.f;
        float vx1 = ((unsigned)x1 < (unsigned)RES) ? 1.f : 0.f;
        float vy0 = ((unsigned)y0 < (unsigned)RES) ? 1.f : 0.f;
        float vy1 = ((unsigned)y1 < (unsigned)RES) ? 1.f : 0.f;
        float w00 = (1.f - wx) * (1.f - wy) * vx0 * vy0;
        float w01 = wx * (1.f - wy) * vx1 * vy0;
        float w10 = (1.f - wx) * wy * vx0 * vy1;
        float w11 = wx * wy * vx1 * vy1;
        int xc0 = iclamp(x0, 0, RES - 1), xc1 = iclamp(x1, 0, RES - 1);
        int yc0 = iclamp(y0, 0, RES - 1), yc1 = iclamp(y1, 0, RES - 1);

        float ivy = fmaf(VY[i], 256.f, 255.5f);
        float fvy = floorf(ivy);
        float wv = ivy - fvy;
        int yv0 = (int)fvy, yv1 = yv0 + 1;
        float w0v = (1.f - wv) * (((unsigned)yv0 < (unsigned)RES) ? 1.f : 0.f);
        float w1v = wv * (((unsigned)yv1 < (unsigned)RES) ? 1.f : 0.f);
        int yv0c = iclamp(yv0, 0, RES - 1), yv1c = iclamp(yv1, 0, RES - 1);

        const float* m0 = Cm + (size_t)i * COUT * RES * RES;
        const float* v0 = Cv + (size_t)i * COUT * RES;
        for (int c = 0; c < COUT; ++c) {
            const float* mc = m0 + (size_t)c * RES * RES;
            float g = fmaf(w11, mc[yc1 * RES + xc1],
                      fmaf(w10, mc[yc1 * RES + xc0],
                      fmaf(w01, mc[yc0 * RES + xc1], w00 * mc[yc0 * RES + xc0])));
            float v = fmaf(w1v, v0[c * RES + yv1c], w0v * v0[c * RES + yv0c]);
            outp[(size_t)p * 96 + i * 32 + c] = g * v;
        }
    }
}

extern "C" void kernel_launch(void* const* d_in, const int* in_sizes, int n_in,
                              void* d_out, int out_size, void* d_ws, size_t ws_size,
                              hipStream_t stream) {
    const float* x  = (const float*)d_in[0];
    const float* Cm = (const float*)d_in[1];
    const float* Cv = (const float*)d_in[2];
    float* out = (float*)d_out;
    int N = in_sizes[0] / 3;

    const size_t needed = (MATT_ELEMS + VEC_ELEMS) * sizeof(float);
    if (ws_size >= needed) {
        float* wsM = (float*)d_ws;
        float* wsV = wsM + MATT_ELEMS;
        k_transpose_mat<<<3 * RES * (RES / 64), 256, 0, stream>>>(Cm, wsM);
        k_transpose_vec<<<(VEC_ELEMS + 255) / 256, 256, 0, stream>>>(Cv, wsV);
        (void)hipFuncSetAttribute((const void*)k_vm_main,
                                  hipFuncAttributeMaxDynamicSharedMemorySize,
                                  (int)(VEC_ELEMS * sizeof(float)));
        int blocks = (N + 1023) / 1024;
        if (blocks > MAIN_BLOCKS) blocks = MAIN_BLOCKS;
        if (blocks < 1) blocks = 1;
        k_vm_main<<<blocks, 1024, VEC_ELEMS * sizeof(float), stream>>>(
            x, wsM, wsV, out, N);
    } else {
        k_vm_direct<<<(N + 255) / 256, 256, 0, stream>>>(x, Cm, Cv, out, N);
    }
}